// Model_1_8675833938328
// MI455X (gfx1250) — compile-verified
//
#include <hip/hip_runtime.h>
#include <hip/hip_bf16.h>

typedef __attribute__((ext_vector_type(2))) float v2f;
typedef __attribute__((ext_vector_type(8))) float v8f;

#define N_NODES  100000
#define N_EDGES  1600000
#define DIM      64
#define N_GRAPHS 256

// ---------------------------------------------------------------------------
// Zero-fill (graph-capture-safe alternative to hipMemsetAsync)
// ---------------------------------------------------------------------------
__global__ void zero_f32(float* __restrict__ p, int n) {
    int i = blockIdx.x * blockDim.x + threadIdx.x;
    if (i < n) p[i] = 0.0f;
}

// ---------------------------------------------------------------------------
// Edge scatter-add: aggr[dst[e], :] += x[src[e], :]
// 16 threads per edge, each handles 4 contiguous features (float4 gather,
// 4x global_atomic_add_f32 scatter).
// ---------------------------------------------------------------------------
__global__ void scatter_add_rows(const float* __restrict__ x,
                                 const int*   __restrict__ src,
                                 const int*   __restrict__ dst,
                                 float*       __restrict__ aggr) {
    int gid = blockIdx.x * blockDim.x + threadIdx.x;
    int e   = gid >> 4;
    if (e >= N_EDGES) return;
    int f = (gid & 15) << 2;
    int s = src[e];
    int d = dst[e];
    const float4 v = *reinterpret_cast<const float4*>(x + (size_t)s * DIM + f);
    float* p = aggr + (size_t)d * DIM + f;
    unsafeAtomicAdd(p + 0, v.x);
    unsafeAtomicAdd(p + 1, v.y);
    unsafeAtomicAdd(p + 2, v.z);
    unsafeAtomicAdd(p + 3, v.w);
}

// ---------------------------------------------------------------------------
// Fused dual GEMM via V_WMMA_F32_16X16X4_F32 (wave32):
//   out[r, :] = A[r, :] @ W1 + bias  (+ B[r, :] @ W2 if DUAL)  (+ReLU if RELU)
// A,B: [rows, 64] row-major; W1,W2: [64, 64] row-major (K x N); bias: [64].
// Block = 256 threads = 8 waves = 2 row-tiles x 4 col-tiles (32 rows / block).
// rows must be a multiple of 32 (100000 and 256 both are).
//
// Fragment layouts (ISA 7.12.2, 32-bit, wave32):
//   A 16x4 : v[0] = A[m0+(lane&15)][k0 + 2*(lane>>4)],     v[1] = ... +1
//   B 4x16 : v[0] = W[(k0 + 2*(lane>>4))*64 + n0+(lane&15)], v[1] = next K row
//   C/D    : vgpr r, row = m0 + r + 8*(lane>>4), col = n0 + (lane&15)
// ---------------------------------------------------------------------------
template <bool DUAL, bool RELU>
__global__ void gemm64_wmma(const float* __restrict__ A,
                            const float* __restrict__ W1,
                            const float* __restrict__ bias,
                            const float* __restrict__ B,
                            const float* __restrict__ W2,
                            float*       __restrict__ out) {
    __shared__ float sW1[DIM * DIM];
    __shared__ float sW2[DIM * DIM];
    for (int i = threadIdx.x; i < DIM * DIM; i += blockDim.x) {
        sW1[i] = W1[i];
        sW2[i] = DUAL ? W2[i] : 0.0f;
    }
    __syncthreads();

    const int lane    = threadIdx.x & 31;
    const int wave    = threadIdx.x >> 5;
    const int rowtile = blockIdx.x * 2 + (wave >> 2);
    const int coltile = wave & 3;
    const int m0   = rowtile * 16;
    const int n0   = coltile * 16;
    const int l15  = lane & 15;
    const int kOff = (lane >> 4) << 1;   // 0 for lanes 0-15, 2 for lanes 16-31
    const int rA   = m0 + l15;           // A-fragment row for this lane
    const int col  = n0 + l15;           // B/C/D column for this lane

    v8f acc = {};
#pragma unroll
    for (int k0 = 0; k0 < DIM; k0 += 4) {
        v2f a;
        {
            const float* p = A + (size_t)rA * DIM + (k0 + kOff);
            a.x = p[0]; a.y = p[1];
        }
        v2f b;
        b.x = sW1[(k0 + kOff) * DIM + col];
        b.y = sW1[(k0 + kOff + 1) * DIM + col];
        acc = __builtin_amdgcn_wmma_f32_16x16x4_f32(
            /*neg_a=*/false, a, /*neg_b=*/false, b,
            /*c_mod=*/(short)0, acc, /*reuse_a=*/false, /*reuse_b=*/false);
        if (DUAL) {
            v2f a2;
            {
                const float* p = B + (size_t)rA * DIM + (k0 + kOff);
                a2.x = p[0]; a2.y = p[1];
            }
            v2f b2;
            b2.x = sW2[(k0 + kOff) * DIM + col];
            b2.y = sW2[(k0 + kOff + 1) * DIM + col];
            acc = __builtin_amdgcn_wmma_f32_16x16x4_f32(
                false, a2, false, b2, (short)0, acc, false, false);
        }
    }

    const float bv    = bias[col];
    const int   rbase = m0 + ((lane >> 4) << 3);
#pragma unroll
    for (int r = 0; r < 8; ++r) {
        float v = acc[r] + bv;
        if (RELU) v = fmaxf(v, 0.0f);
        out[(size_t)(rbase + r) * DIM + col] = v;
    }
}

// ---------------------------------------------------------------------------
// Mean pool: sums[batch[n], :] += h[n, :]; counts[batch[n]] += 1
// ---------------------------------------------------------------------------
__global__ void pool_accum(const float* __restrict__ h,
                           const int*   __restrict__ batch,
                           float*       __restrict__ sums,
                           float*       __restrict__ counts) {
    int gid  = blockIdx.x * blockDim.x + threadIdx.x;
    int node = gid >> 4;
    if (node >= N_NODES) return;
    int f = (gid & 15) << 2;
    int g = batch[node];
    const float4 v = *reinterpret_cast<const float4*>(h + (size_t)node * DIM + f);
    float* p = sums + (size_t)g * DIM + f;
    unsafeAtomicAdd(p + 0, v.x);
    unsafeAtomicAdd(p + 1, v.y);
    unsafeAtomicAdd(p + 2, v.z);
    unsafeAtomicAdd(p + 3, v.w);
    if ((gid & 15) == 0) unsafeAtomicAdd(&counts[g], 1.0f);
}

__global__ void pool_div(const float* __restrict__ sums,
                         const float* __restrict__ counts,
                         float*       __restrict__ pooled) {
    int i = blockIdx.x * blockDim.x + threadIdx.x;
    if (i < N_GRAPHS * DIM) {
        float c = counts[i >> 6];
        pooled[i] = sums[i] / fmaxf(c, 1.0f);
    }
}

// ---------------------------------------------------------------------------
// Orchestration
// ---------------------------------------------------------------------------
extern "C" void kernel_launch(void* const* d_in, const int* in_sizes, int n_in,
                              void* d_out, int out_size, void* d_ws, size_t ws_size,
                              hipStream_t stream) {
    const float* x     = (const float*)d_in[0];
    const int*   edge  = (const int*)  d_in[1];
    const int*   src   = edge;
    const int*   dst   = edge + N_EDGES;
    const int*   batch = (const int*)  d_in[2];
    const float* Wrel1 = (const float*)d_in[3];
    const float* brel1 = (const float*)d_in[4];
    const float* Wroot1= (const float*)d_in[5];
    const float* Wrel2 = (const float*)d_in[6];
    const float* brel2 = (const float*)d_in[7];
    const float* Wroot2= (const float*)d_in[8];
    const float* Wrel3 = (const float*)d_in[9];
    const float* brel3 = (const float*)d_in[10];
    const float* Wroot3= (const float*)d_in[11];
    const float* Wlin  = (const float*)d_in[12];
    const float* blin  = (const float*)d_in[13];
    float*       outp  = (float*)d_out;

    // Workspace layout (all offsets 16B aligned)
    char*  ws     = (char*)d_ws;
    float* aggr   = (float*)(ws + 0);            // 100000*64 f32 = 25.6 MB
    float* hA     = (float*)(ws + 25600000);     // 25.6 MB
    float* hB     = (float*)(ws + 51200000);     // 25.6 MB
    float* sums   = (float*)(ws + 76800000);     // 256*64 f32
    float* counts = (float*)(ws + 76865536);     // 256 f32
    float* pooled = (float*)(ws + 76866560);     // 256*64 f32

    const int NF = N_NODES * DIM;                // 6,400,000
    const int ZB = (NF + 255) / 256;
    const int SB = (N_EDGES * 16) / 256;         // 100000 blocks, exact
    const int GB = N_NODES / 32;                 // 3125 blocks, exact
    const int PB = (N_NODES * 16) / 256;         // 6250 blocks, exact

    // ---- Layer 1: h = relu(aggr(x)@Wrel1 + brel1 + x@Wroot1)
    zero_f32<<<ZB, 256, 0, stream>>>(aggr, NF);
    scatter_add_rows<<<SB, 256, 0, stream>>>(x, src, dst, aggr);
    gemm64_wmma<true, true><<<GB, 256, 0, stream>>>(aggr, Wrel1, brel1, x, Wroot1, hA);

    // ---- Layer 2
    zero_f32<<<ZB, 256, 0, stream>>>(aggr, NF);
    scatter_add_rows<<<SB, 256, 0, stream>>>(hA, src, dst, aggr);
    gemm64_wmma<true, true><<<GB, 256, 0, stream>>>(aggr, Wrel2, brel2, hA, Wroot2, hB);

    // ---- Layer 3 (no relu)
    zero_f32<<<ZB, 256, 0, stream>>>(aggr, NF);
    scatter_add_rows<<<SB, 256, 0, stream>>>(hB, src, dst, aggr);
    gemm64_wmma<true, false><<<GB, 256, 0, stream>>>(aggr, Wrel3, brel3, hB, Wroot3, hA);

    // ---- Global mean pool (sums + counts are contiguous: zero both at once)
    zero_f32<<<(N_GRAPHS * DIM + N_GRAPHS + 255) / 256, 256, 0, stream>>>(sums, N_GRAPHS * DIM + N_GRAPHS);
    pool_accum<<<PB, 256, 0, stream>>>(hA, batch, sums, counts);
    pool_div<<<(N_GRAPHS * DIM + 255) / 256, 256, 0, stream>>>(sums, counts, pooled);

    // ---- Final linear: out = pooled @ Wlin + blin   (256 rows = 8 blocks)
    gemm64_wmma<false, false><<<N_GRAPHS / 32, 256, 0, stream>>>(
        pooled, Wlin, blin, nullptr, nullptr, outp);
}